// SoftmaxWeightedSum_16355235463569
// MI455X (gfx1250) — compile-verified
//
#include <hip/hip_runtime.h>
#include <hip/hip_bf16.h>

// Masked softmax (T=8192) + weighted sum against value[B,T,64] -> out[B,1,64].
// HBM-bound (261 MB @ 23.3 TB/s ~ 11us): single pass over `value`, partial
// (flash-style) softmax per chunk, f32 WMMA 16x16x4 for the weighted sum so
// we keep full f32 precision while exercising the CDNA5 matrix pipe.

#define NEG_INF (-4294967295.0f) /* -2^32+1, matches reference */

typedef __attribute__((ext_vector_type(2))) float v2f;
typedef __attribute__((ext_vector_type(8))) float v8f;

#define B_DIM   128
#define T_DIM   8192
#define U_DIM   64
#define NCHUNK  4
#define TCHUNK  (T_DIM / NCHUNK)     /* 2048 t per workgroup */
#define THREADS 256                  /* 8 wave32 */
#define NWAVE   (THREADS / 32)
#define WT      (TCHUNK / NWAVE)     /* 256 t per wave */

__global__ __launch_bounds__(THREADS)
void SoftmaxWeightedSum_partial(const float* __restrict__ align,
                                const float* __restrict__ value,
                                const unsigned char* __restrict__ masks,
                                float* __restrict__ Pm,
                                float* __restrict__ Ps,
                                float* __restrict__ Pacc)
{
    __shared__ float p_lds[TCHUNK];          // scores, then softmax numerators
    __shared__ float red[THREADS];           // block reductions
    __shared__ float wacc[NWAVE * U_DIM];    // per-wave partial outputs

    const int chunk = blockIdx.x;
    const int b     = blockIdx.y;
    const int tid   = threadIdx.x;
    const int lane  = tid & 31;
    const int w     = tid >> 5;
    const int half  = lane >> 4;             // 0: lanes 0-15, 1: lanes 16-31
    const int col   = lane & 15;

    const int    tBase   = chunk * TCHUNK;
    const size_t rowBase = (size_t)b * T_DIM;

    // ---- Phase 1: masked scores -> LDS, block max -------------------------
    float m_local = NEG_INF;
    #pragma unroll
    for (int i = 0; i < TCHUNK / THREADS; ++i) {
        int   idx = i * THREADS + tid;
        int   t   = tBase + idx;
        float a   = align[rowBase + t];
        float sc  = masks[rowBase + t] ? a : NEG_INF;
        p_lds[idx] = sc;
        m_local    = fmaxf(m_local, sc);
    }
    red[tid] = m_local;
    __syncthreads();
    for (int s = THREADS / 2; s > 0; s >>= 1) {
        if (tid < s) red[tid] = fmaxf(red[tid], red[tid + s]);
        __syncthreads();
    }
    const float m = red[0];
    __syncthreads();

    // ---- Phase 2: p = exp(sc - m) (masked -> exactly 0), block sum --------
    float s_local = 0.f;
    #pragma unroll
    for (int i = 0; i < TCHUNK / THREADS; ++i) {
        int   idx = i * THREADS + tid;
        float p   = __expf(p_lds[idx] - m);   // exp(NEG_INF - m) underflows to 0
        p_lds[idx] = p;
        s_local   += p;
    }
    red[tid] = s_local;
    __syncthreads();
    for (int s = THREADS / 2; s > 0; s >>= 1) {
        if (tid < s) red[tid] += red[tid + s];
        __syncthreads();
    }
    const float s_chunk = red[0];

    // ---- Phase 3: WMMA f32 16x16x4 accumulation over this wave's 256 t ----
    // A (16x4 f32, ISA 7.12.2): VGPR0 = K=2*half, VGPR1 = K=2*half+1, all M
    // rows identical (= softmax weights) so every C row equals the answer.
    // B (4x16 f32): lane = N column, VGPR e + lane-half select K = 2*half+e.
    v8f acc0 = {}, acc1 = {}, acc2 = {}, acc3 = {};
    const int    tw    = w * WT;
    const float* vbase = value + (rowBase + tBase + tw) * (size_t)U_DIM;

    for (int kk = 0; kk < WT / 4; ++kk) {
        const int tloc = kk * 4;
        v2f a;
        a[0] = p_lds[tw + tloc + 2 * half];
        a[1] = p_lds[tw + tloc + 2 * half + 1];

        const float* vrow = vbase + (size_t)(tloc + 2 * half) * U_DIM + col;
        if (kk < (WT / 4 - 8))
            __builtin_prefetch(vrow + 32 * U_DIM, 0, 0);  // global_prefetch_b8

        v2f b0, b1, b2, b3;
        b0[0] = vrow[0];   b0[1] = vrow[U_DIM];
        b1[0] = vrow[16];  b1[1] = vrow[U_DIM + 16];
        b2[0] = vrow[32];  b2[1] = vrow[U_DIM + 32];
        b3[0] = vrow[48];  b3[1] = vrow[U_DIM + 48];

        acc0 = __builtin_amdgcn_wmma_f32_16x16x4_f32(false, a, false, b0, (short)0, acc0, false, false);
        acc1 = __builtin_amdgcn_wmma_f32_16x16x4_f32(false, a, false, b1, (short)0, acc1, false, false);
        acc2 = __builtin_amdgcn_wmma_f32_16x16x4_f32(false, a, false, b2, (short)0, acc2, false, false);
        acc3 = __builtin_amdgcn_wmma_f32_16x16x4_f32(false, a, false, b3, (short)0, acc3, false, false);
    }

    // C/D layout: VGPR0, lanes 0-15 hold row M=0, N=lane.
    if (lane < 16) {
        wacc[w * U_DIM +      lane] = acc0[0];
        wacc[w * U_DIM + 16 + lane] = acc1[0];
        wacc[w * U_DIM + 32 + lane] = acc2[0];
        wacc[w * U_DIM + 48 + lane] = acc3[0];
    }
    __syncthreads();

    const int r = b * NCHUNK + chunk;
    if (tid == 0) { Pm[r] = m; Ps[r] = s_chunk; }
    if (tid < U_DIM) {
        float a = 0.f;
        #pragma unroll
        for (int ww = 0; ww < NWAVE; ++ww) a += wacc[ww * U_DIM + tid];
        Pacc[(size_t)r * U_DIM + tid] = a;
    }
}

// ---- Combine NCHUNK partial-softmax records per batch ----------------------
__global__ __launch_bounds__(U_DIM)
void SoftmaxWeightedSum_combine(const float* __restrict__ Pm,
                                const float* __restrict__ Ps,
                                const float* __restrict__ Pacc,
                                float* __restrict__ out)
{
    const int b = blockIdx.x;
    const int u = threadIdx.x;

    float M = NEG_INF;
    #pragma unroll
    for (int c = 0; c < NCHUNK; ++c) M = fmaxf(M, Pm[b * NCHUNK + c]);

    float S = 0.f, A = 0.f;
    #pragma unroll
    for (int c = 0; c < NCHUNK; ++c) {
        const int   r = b * NCHUNK + c;
        const float f = __expf(Pm[r] - M);
        S += Ps[r] * f;
        A += Pacc[(size_t)r * U_DIM + u] * f;
    }
    out[b * U_DIM + u] = A / S;
}

extern "C" void kernel_launch(void* const* d_in, const int* in_sizes, int n_in,
                              void* d_out, int out_size, void* d_ws, size_t ws_size,
                              hipStream_t stream) {
    const float*         align = (const float*)d_in[0];          // [B,1,T] f32
    const float*         value = (const float*)d_in[1];          // [B,T,U] f32
    const unsigned char* masks = (const unsigned char*)d_in[2];  // [B,1,T] bool
    float*               out   = (float*)d_out;                  // [B,1,U] f32

    float* Pm   = (float*)d_ws;                   // [B*NCHUNK]
    float* Ps   = Pm + B_DIM * NCHUNK;            // [B*NCHUNK]
    float* Pacc = Ps + B_DIM * NCHUNK;            // [B*NCHUNK, U]

    dim3 grid(NCHUNK, B_DIM);
    SoftmaxWeightedSum_partial<<<grid, THREADS, 0, stream>>>(align, value, masks, Pm, Ps, Pacc);
    SoftmaxWeightedSum_combine<<<B_DIM, U_DIM, 0, stream>>>(Pm, Ps, Pacc, out);
}